// FlowMemory_25709674234631
// MI455X (gfx1250) — compile-verified
//
#include <hip/hip_runtime.h>

typedef __attribute__((ext_vector_type(2))) float v2f;
typedef __attribute__((ext_vector_type(8))) float v8f;

#define N_DIM   4096
#define THREE_D 192
#define DMEM    64
#define TSTEPS  16
#define SHP     68   // LDS row pitch (dwords): stride-4 banks -> conflict-free b64 frags

// ---------------------------------------------------------------------------
// Kernel A: GI(t, j, m) = sum_k W_ih(j, k) * F(t, m, k) + b_ih(j)
// Computed as D = W_tile(16j x 4k) x F^T(4k x 16m) so the C layout puts m
// across lanes -> coalesced 64B store segments into the (t, j, m) layout.
// One wave: 16 m-rows x all 192 j-cols (12 tiles), K swept 4 at a time.
// ---------------------------------------------------------------------------
__global__ __launch_bounds__(128) void gi_gemm_kernel(
    const float* __restrict__ F,    // (16, 4096, 4096)
    const float* __restrict__ Wih,  // (192, 4096)
    const float* __restrict__ bih,  // (192,)
    float* __restrict__ GI)         // (16, 192, 4096) in d_ws
{
    const int lane  = threadIdx.x & 31;
    const int wave  = threadIdx.x >> 5;
    const int row0  = (blockIdx.x * 4 + wave) * 16;   // global F row tile
    const int l15   = lane & 15;
    const int khalf = (lane >> 4) << 1;               // 0 or 2 (frag K split)

    // B fragment (F^T tile): lane holds F[row0+l15][k + khalf + {0,1}]
    const float* fp = F + (size_t)(row0 + l15) * N_DIM + khalf;
    // A fragment (W tile): lane holds Wih[jt*16+l15][k + khalf + {0,1}]
    const float* wp = Wih + (size_t)l15 * N_DIM + khalf;

    v8f acc[12] = {};

    for (int kb = 0; kb < N_DIM / 4; ++kb) {
        v2f b = *(const v2f*)fp;
        fp += 4;
#pragma unroll
        for (int j = 0; j < 12; ++j) {
            v2f a = *(const v2f*)(wp + (size_t)j * 16 * N_DIM);
            acc[j] = __builtin_amdgcn_wmma_f32_16x16x4_f32(
                false, a, false, b, (short)0, acc[j], false, false);
        }
        wp += 4;
    }

    // C layout: VGPR v -> j = jt*16 + v + 8*(lane>>4); lanes&15 -> m (coalesced)
    const int t    = row0 >> 12;
    const int mloc = (row0 & (N_DIM - 1)) + l15;
    const int jadd = (lane >> 4) * 8;
#pragma unroll
    for (int jt = 0; jt < 12; ++jt) {
        const v8f bb = *(const v8f*)(bih + jt * 16 + jadd);  // 32B aligned
#pragma unroll
        for (int v = 0; v < 8; ++v) {
            const int j = jt * 16 + jadd + v;
            GI[((size_t)t * THREE_D + j) * N_DIM + mloc] = acc[jt][v] + bb[v];
        }
    }
}

// ---------------------------------------------------------------------------
// Prep: swizzle W_hh / W_proj into fragment-ordered b64 tables (read as
// coalesced global_load_b64 by every wave of the recurrence kernel; 64 KB
// total -> resident in every WGP$).
// Table element e = (k0*NT + jt)*32 + lane holds W[jt*16+(l&15)][k0*4+2*(l>>4)+{0,1}]
// ---------------------------------------------------------------------------
__global__ __launch_bounds__(256) void frag_prep_kernel(
    const float* __restrict__ Whh, const float* __restrict__ Wproj,
    float* __restrict__ WhhFrag, float* __restrict__ WprojFrag)
{
    const int tid = threadIdx.x;
    for (int e = tid; e < 16 * 12 * 32; e += 256) {
        const int l = e & 31, jt = (e >> 5) % 12, k0 = e / (12 * 32);
        ((v2f*)WhhFrag)[e] =
            *(const v2f*)(Whh + (size_t)(jt * 16 + (l & 15)) * DMEM + k0 * 4 + 2 * (l >> 4));
    }
    for (int e = tid; e < 16 * 4 * 32; e += 256) {
        const int l = e & 31, jt = (e >> 5) % 4, k0 = e / (4 * 32);
        ((v2f*)WprojFrag)[e] =
            *(const v2f*)(Wproj + (size_t)(jt * 16 + (l & 15)) * DMEM + k0 * 4 + 2 * (l >> 4));
    }
}

// ---------------------------------------------------------------------------
// Kernel B: WMMA GRU recurrence. One wave owns 16 rows for all 16 steps.
// Per step: GH = H @ Whh^T + bhh (192 WMMAs, A-frags from LDS h-tile),
// elementwise gates on C-tiles (r/z/n tiles share lane->(m,j) mapping,
// h_old kept in registers), write new h to LDS, then E = H @ Wproj^T + bproj
// (64 WMMAs) stored coalesced. No cross-wave data -> single-wave workgroups.
// ---------------------------------------------------------------------------
__global__ __launch_bounds__(32) void gru_wmma_kernel(
    const float* __restrict__ GI,        // (16, 192, 4096)
    const float* __restrict__ mem0,      // (4096, 64)
    const float* __restrict__ WhhFrag,   // fragment table
    const float* __restrict__ bhh,       // (192,)
    const float* __restrict__ WprojFrag, // fragment table
    const float* __restrict__ bproj,     // (64,)
    float* __restrict__ out)             // (16, 4096, 64)
{
    __shared__ float sH[16 * SHP];
    const int lane = threadIdx.x;
    const int l15  = lane & 15;
    const int lh   = lane >> 4;
    const int row0 = blockIdx.x * 16;

    float bh[12], bp[4];
#pragma unroll
    for (int jg = 0; jg < 12; ++jg) bh[jg] = bhh[jg * 16 + l15];
#pragma unroll
    for (int jt = 0; jt < 4; ++jt) bp[jt] = bproj[jt * 16 + l15];

    // h tiles in C layout: element (m = v + 8*lh, j = jt*16 + l15)
    v8f hreg[4];
#pragma unroll
    for (int jt = 0; jt < 4; ++jt) {
#pragma unroll
        for (int v = 0; v < 8; ++v) {
            const float hv = mem0[(size_t)(row0 + v + 8 * lh) * DMEM + jt * 16 + l15];
            hreg[jt][v] = hv;
            sH[(v + 8 * lh) * SHP + jt * 16 + l15] = hv;
        }
    }
    __syncthreads();

    const v2f* WhF = (const v2f*)WhhFrag;
    const v2f* WpF = (const v2f*)WprojFrag;
    const int aoff = l15 * SHP + 2 * lh;   // A-frag LDS dword offset

    for (int t = 0; t < TSTEPS; ++t) {
        // ---- GH = H @ Whh^T + bhh ----
        v8f acc[12];
#pragma unroll
        for (int jg = 0; jg < 12; ++jg) {
#pragma unroll
            for (int v = 0; v < 8; ++v) acc[jg][v] = bh[jg];
        }
#pragma unroll 4
        for (int k0 = 0; k0 < 16; ++k0) {
            v2f a = *(const v2f*)(&sH[aoff + k0 * 4]);   // ds_load_b64, conflict-free
#pragma unroll
            for (int jg = 0; jg < 12; ++jg) {
                v2f b = WhF[(k0 * 12 + jg) * 32 + lane]; // coalesced b64, WGP$-hot
                acc[jg] = __builtin_amdgcn_wmma_f32_16x16x4_f32(
                    false, a, false, b, (short)0, acc[jg], false, false);
            }
        }

        // ---- gates (elementwise on C tiles; jt, jt+4, jt+8 = r, z, n) ----
        const float* gbase = GI + ((size_t)t * THREE_D + l15) * N_DIM + row0 + 8 * lh;
#pragma unroll
        for (int jt = 0; jt < 4; ++jt) {
            const float* g = gbase + (size_t)jt * 16 * N_DIM;
            v8f gr = *(const v8f*)(g);                          // 32B-aligned
            v8f gz = *(const v8f*)(g + (size_t)64 * N_DIM);
            v8f gn = *(const v8f*)(g + (size_t)128 * N_DIM);
            v8f hnew;
#pragma unroll
            for (int v = 0; v < 8; ++v) {
                const float r  = 1.0f / (1.0f + __expf(-(gr[v] + acc[jt][v])));
                const float z  = 1.0f / (1.0f + __expf(-(gz[v] + acc[4 + jt][v])));
                const float nn = tanhf(gn[v] + r * acc[8 + jt][v]);
                hnew[v] = (1.0f - z) * nn + z * hreg[jt][v];
            }
            hreg[jt] = hnew;
#pragma unroll
            for (int v = 0; v < 8; ++v)       // conflict-free banks (halves disjoint)
                sH[(v + 8 * lh) * SHP + jt * 16 + l15] = hnew[v];
        }
        __syncthreads();   // single-wave WG: compiler fence + dscnt ordering

        // ---- E = newH @ Wproj^T + bproj ----
        v8f eacc[4];
#pragma unroll
        for (int jt = 0; jt < 4; ++jt) {
#pragma unroll
            for (int v = 0; v < 8; ++v) eacc[jt][v] = bp[jt];
        }
#pragma unroll 4
        for (int k0 = 0; k0 < 16; ++k0) {
            v2f a = *(const v2f*)(&sH[aoff + k0 * 4]);
#pragma unroll
            for (int jt = 0; jt < 4; ++jt) {
                v2f b = WpF[(k0 * 4 + jt) * 32 + lane];
                eacc[jt] = __builtin_amdgcn_wmma_f32_16x16x4_f32(
                    false, a, false, b, (short)0, eacc[jt], false, false);
            }
        }

        // store: element (m = row0 + v + 8*lh, o = jt*16 + l15) -> coalesced 64B
        float* ob = out + ((size_t)t * N_DIM + row0 + 8 * lh) * DMEM + l15;
#pragma unroll
        for (int jt = 0; jt < 4; ++jt) {
#pragma unroll
            for (int v = 0; v < 8; ++v)
                ob[(size_t)v * DMEM + jt * 16] = eacc[jt][v];
        }
        __syncthreads();
    }
}

extern "C" void kernel_launch(void* const* d_in, const int* in_sizes, int n_in,
                              void* d_out, int out_size, void* d_ws, size_t ws_size,
                              hipStream_t stream) {
    const float* F     = (const float*)d_in[0];  // (16, 4096, 4096)
    const float* mem0  = (const float*)d_in[1];  // (4096, 64)
    const float* Wih   = (const float*)d_in[2];  // (192, 4096)
    const float* Whh   = (const float*)d_in[3];  // (192, 64)
    const float* bih   = (const float*)d_in[4];  // (192,)
    const float* bhh   = (const float*)d_in[5];  // (192,)
    const float* Wproj = (const float*)d_in[6];  // (64, 64)
    const float* bproj = (const float*)d_in[7];  // (64,)
    float* out = (float*)d_out;                  // (16, 4096, 64)

    float* GI        = (float*)d_ws;                              // 50.3 MB
    float* WhhFrag   = GI + (size_t)TSTEPS * THREE_D * N_DIM;     // 48 KB
    float* WprojFrag = WhhFrag + THREE_D * DMEM;                  // 16 KB

    frag_prep_kernel<<<1, 256, 0, stream>>>(Whh, Wproj, WhhFrag, WprojFrag);
    gi_gemm_kernel<<<1024, 128, 0, stream>>>(F, Wih, bih, GI);
    gru_wmma_kernel<<<N_DIM / 16, 32, 0, stream>>>(GI, mem0, WhhFrag, bhh,
                                                   WprojFrag, bproj, out);
}